// ShellProvider_1872605741128
// MI455X (gfx1250) — compile-verified
//
#include <hip/hip_runtime.h>
#include <stdint.h>

// Reference shapes: B=8 molecules, A=8192 atoms, K=128 candidate neighbors.
// Memory-bound gather + wave32 stable stream-compaction.
//
// CDNA5-specific paths used:
//  - global_load_async_to_lds_b128 (ASYNCcnt DMA copy) to stage R[b] (96 KB)
//    into LDS per workgroup, so the random 12-byte coordinate gathers hit LDS
//    banks instead of HBM/L2. Unrolled into a 24-issue burst per thread-slice.
//  - s_wait_asynccnt 0 to drain the async copies before the barrier.
//  - Non-temporal (TH=NT) hints on the read-once index/mask streams and all
//    write-once outputs, so L2 stays reserved for the reused coordinate tables.
//  - wave32 ballot/popc compaction (one wave per row of K=128).

#define CUTOFF_F 5.0f

constexpr int A_ATOMS        = 8192;
constexpr int K_NB           = 128;
constexpr int THREADS        = 256;                            // 8 waves (wave32)
constexpr int WAVES_PER_BLK  = THREADS / 32;                   // 8
constexpr int BLOCKS_PER_MOL = 64;
constexpr int ROWS_PER_BLOCK = A_ATOMS / BLOCKS_PER_MOL;       // 128
constexpr int ROWS_PER_WAVE  = ROWS_PER_BLOCK / WAVES_PER_BLK; // 16

__global__ __launch_bounds__(THREADS)
void neighbor_compact_kernel(const float* __restrict__ R,   // [B,A,3]
                             const int*   __restrict__ N,   // [B,A,K]
                             const float* __restrict__ NM,  // [B,A,K]
                             float* __restrict__ outD,      // [B,A,K]
                             float* __restrict__ outV,      // [B,A,K,3]
                             float* __restrict__ outN,      // [B,A,K] (int value as float)
                             float* __restrict__ outNM)     // [B,A,K]
{
    __shared__ float Rs[A_ATOMS * 3];   // 96 KB, packed xyz per atom

    const int b    = blockIdx.x / BLOCKS_PER_MOL;
    const int blk  = blockIdx.x % BLOCKS_PER_MOL;
    const int tid  = threadIdx.x;
    const int lane = tid & 31;
    const int wave = tid >> 5;

    // ---- Stage R[b] into LDS via CDNA5 async global->LDS DMA (16B granules) ----
    // GRANULES divides exactly by THREADS -> constant-trip unrolled burst, no
    // per-iteration EXEC masking.
    const float* Rb = R + (size_t)b * (A_ATOMS * 3);
    const uint32_t ldsBase = (uint32_t)(uintptr_t)(&Rs[0]); // low 32 bits = LDS offset
    constexpr int GRANULES = (A_ATOMS * 3 * 4) / 16;        // 6144 x 16B
    constexpr int COPY_ITERS = GRANULES / THREADS;          // 24
    #pragma unroll
    for (int i = 0; i < COPY_ITERS; ++i) {
        const int t = i * THREADS + tid;
        const uint32_t ldsAddr = ldsBase + (uint32_t)t * 16u;
        const float* g = Rb + t * 4;
        asm volatile("global_load_async_to_lds_b128 %0, %1, off"
                     :: "v"(ldsAddr), "v"(g)
                     : "memory");
    }
    asm volatile("s_wait_asynccnt 0" ::: "memory");
    __syncthreads();

    // ---- One wave per row; K=128 processed as 4 chunks of 32 lanes ----
    const int row0 = blk * ROWS_PER_BLOCK + wave * ROWS_PER_WAVE;

    for (int r = 0; r < ROWS_PER_WAVE; ++r) {
        const int a = row0 + r;
        const size_t rowBase = ((size_t)b * A_ATOMS + (size_t)a) * K_NB;

        // broadcast read of this row's own atom coords from LDS
        const float ax = Rs[3 * a + 0];
        const float ay = Rs[3 * a + 1];
        const float az = Rs[3 * a + 2];

        int base = 0; // running count of valid entries (stable compaction base)
        #pragma unroll
        for (int c = 0; c < K_NB / 32; ++c) {
            const int k = c * 32 + lane;
            // read-once streams: non-temporal loads (TH=NT)
            const int   n  = __builtin_nontemporal_load(&N [rowBase + k]);
            const float nm = __builtin_nontemporal_load(&NM[rowBase + k]);

            // random gather from LDS (molecule-local coordinate table)
            const float vx = Rs[3 * n + 0] - ax;
            const float vy = Rs[3 * n + 1] - ay;
            const float vz = Rs[3 * n + 2] - az;
            const float d  = sqrtf(vx * vx + vy * vy + vz * vz);

            const bool valid = (nm != 0.0f) && (d < CUTOFF_F);

            // wave32 stable compaction: exclusive prefix of `valid` in lane order
            const unsigned mbits = __builtin_amdgcn_ballot_w32(valid);
            const int pos = base + __popc(mbits & ((1u << lane) - 1u));

            if (valid) {
                const size_t o = rowBase + (size_t)pos;
                __builtin_nontemporal_store(d,         &outD [o]);
                __builtin_nontemporal_store(vx,        &outV [o * 3 + 0]);
                __builtin_nontemporal_store(vy,        &outV [o * 3 + 1]);
                __builtin_nontemporal_store(vz,        &outV [o * 3 + 2]);
                __builtin_nontemporal_store((float)n,  &outN [o]);
                __builtin_nontemporal_store(1.0f,      &outNM[o]);
            }
            base += __popc(mbits);
        }

        // zero-fill the tail [base, K) — disjoint from scattered valid slots
        for (int k = base + lane; k < K_NB; k += 32) {
            const size_t o = rowBase + (size_t)k;
            __builtin_nontemporal_store(0.0f, &outD [o]);
            __builtin_nontemporal_store(0.0f, &outV [o * 3 + 0]);
            __builtin_nontemporal_store(0.0f, &outV [o * 3 + 1]);
            __builtin_nontemporal_store(0.0f, &outV [o * 3 + 2]);
            __builtin_nontemporal_store(0.0f, &outN [o]);
            __builtin_nontemporal_store(0.0f, &outNM[o]);
        }
    }
}

extern "C" void kernel_launch(void* const* d_in, const int* in_sizes, int n_in,
                              void* d_out, int out_size, void* d_ws, size_t ws_size,
                              hipStream_t stream) {
    const float* R  = (const float*)d_in[0];   // [B,A,3] f32
    const int*   N  = (const int*)  d_in[1];   // [B,A,K] i32
    const float* NM = (const float*)d_in[2];   // [B,A,K] f32

    const int BAK = in_sizes[1];               // B*A*K
    const int B   = in_sizes[0] / (A_ATOMS * 3);

    // outputs concatenated flat in return order: D_c, V_c, N_c, NM_c
    float* outD  = (float*)d_out;
    float* outV  = outD + (size_t)BAK;
    float* outN  = outV + (size_t)BAK * 3;
    float* outNM = outN + (size_t)BAK;

    dim3 grid(B * BLOCKS_PER_MOL);
    neighbor_compact_kernel<<<grid, THREADS, 0, stream>>>(
        R, N, NM, outD, outV, outN, outNM);
}